// NaiveMHCLayerDynamic_44023414784002
// MI455X (gfx1250) — compile-verified
//
#include <hip/hip_runtime.h>
#include <stdint.h>
#include <stddef.h>

#define B_TOT    8192
#define N_H      4
#define C_DIM    2048
#define NC       8192          // N_H * C_DIM (flattened row length)
#define ROWS     8             // batch rows per block (8*8192*4B = 256KB LDS tile)
#define NTHREADS 256
#define NWAVES   8
#define PC       32            // packed phi columns (24 real + 8 zero pad)
#define EPS_F    1e-5f
#define SINK_IT  20

typedef float v2f __attribute__((ext_vector_type(2)));
typedef float v8f __attribute__((ext_vector_type(8)));
typedef int   v4i __attribute__((ext_vector_type(4)));
typedef __attribute__((address_space(1))) v4i* gv4i_p;   // global int4*
typedef __attribute__((address_space(3))) v4i* lv4i_p;   // LDS int4*

#if defined(__has_builtin)
#  if __has_builtin(__builtin_amdgcn_global_load_async_to_lds_b128)
#    define HAVE_ASYNC_LDS 1
#  endif
#  if __has_builtin(__builtin_amdgcn_s_wait_asynccnt)
#    define HAVE_WAIT_ASYNC 1
#  endif
#endif

__device__ __forceinline__ void async_copy_b128(const float* g, float* l) {
#ifdef HAVE_ASYNC_LDS
  __builtin_amdgcn_global_load_async_to_lds_b128(
      (gv4i_p)(void*)(g), (lv4i_p)(void*)(l), 0, 0);
#else
  *(float4*)l = *(const float4*)g;    // synchronous fallback (still correct)
#endif
}

__device__ __forceinline__ void wait_async_all() {
#ifdef HAVE_ASYNC_LDS
# ifdef HAVE_WAIT_ASYNC
  __builtin_amdgcn_s_wait_asynccnt(0);
# else
  asm volatile("s_wait_asynccnt 0" ::: "memory");
# endif
#endif
}

// ---------------------------------------------------------------------------
// Pack phi_pre(8192x4) | phi_post(8192x4) | phi_res(8192x16) into a WMMA-B
// friendly buffer: for even k, element (k,col) at (k>>1)*64 + col*2, element
// (k+1,col) at +1, so a lane's b64 load yields its {K, K+1} pair.
// Cols 24..31 are zero padding (second 16-wide N tile).
// ---------------------------------------------------------------------------
__global__ void pack_phi_kernel(const float* __restrict__ pre,
                                const float* __restrict__ post,
                                const float* __restrict__ res,
                                float* __restrict__ phiP) {
  int idx = blockIdx.x * blockDim.x + threadIdx.x;   // over NC*PC
  if (idx >= NC * PC) return;
  int k = idx >> 5, col = idx & 31;
  float v = 0.f;
  if (col < 4)        v = pre [k * 4  + col];
  else if (col < 8)   v = post[k * 4  + (col - 4)];
  else if (col < 24)  v = res [k * 16 + (col - 8)];
  phiP[((k >> 1) << 6) + (col << 1) + (k & 1)] = v;
}

// ---------------------------------------------------------------------------
// Fused MHC layer: one block = 8 batch rows, whole tile resident in LDS.
// HBM traffic: read x once (async DMA to LDS), write out once.
// All loops use compile-time trip counts so branches stay scalar (SALU) and
// EXEC is statically all-ones at every WMMA.
// ---------------------------------------------------------------------------
__global__ __launch_bounds__(NTHREADS, 1)
void mhc_fused_kernel(const float* __restrict__ x,
                      const float* __restrict__ wvec,
                      const float* __restrict__ phiP,
                      const float* __restrict__ b_pre,
                      const float* __restrict__ b_post,
                      const float* __restrict__ b_res,
                      const float* __restrict__ a_pre,
                      const float* __restrict__ a_post,
                      const float* __restrict__ a_res,
                      float* __restrict__ outp) {
  __shared__ float xs[ROWS * NC];                // 256 KB  (x tile)
  __shared__ float cpart[NWAVES][16][PC];        // 16 KB   (per-wave WMMA partials)
  __shared__ float dots[16][PC];                 // 2 KB    (reduced projections)
  __shared__ float redbuf[NTHREADS];             // 1 KB
  __shared__ float rms_inv_s[ROWS];
  __shared__ float agg_inv_s[ROWS];
  __shared__ float Hpre_s [ROWS][4];
  __shared__ float Hpost_s[ROWS][4];
  __shared__ float Msink_s[ROWS][16];

  const int t    = threadIdx.x;
  const int wave = t >> 5;
  const int lane = t & 31;
  const int b0   = blockIdx.x * ROWS;

  // ---- Phase 0: async DMA the 8x8192 fp32 tile into LDS -------------------
  {
    const float* gx = x + (size_t)b0 * NC;
    #pragma unroll
    for (int i = 0; i < (ROWS * NC / 4) / NTHREADS; ++i) {   // 64 x b128 / thread
      int f4 = t + i * NTHREADS;
      async_copy_b128(gx + (size_t)f4 * 4, xs + (size_t)f4 * 4);
    }
    wait_async_all();
  }
  __syncthreads();

  // ---- Phase 1a: per-row sum of squares (flat RMS over n*C) ---------------
  {
    const float* xr = xs + (size_t)(t >> 5) * NC + lane;   // 32 threads per row
    float acc = 0.f;
    for (int s = 0; s < NC / 32; ++s) {                    // uniform 256 iters
      float v = xr[s * 32];
      acc = fmaf(v, v, acc);
    }
    redbuf[t] = acc;
  }
  __syncthreads();
  if (t < ROWS) {
    float s = 0.f;
    for (int l = 0; l < 32; ++l) s += redbuf[t * 32 + l];
    rms_inv_s[t] = 1.f / sqrtf(s / (float)NC + EPS_F);
  }
  __syncthreads();

  // ---- Phase 1b: projection GEMM x(8x8192) @ phiP(8192x32) on matrix pipe -
  // A layout (16x4 f32): lanes 0-15 hold M=lane, K={0,1}; lanes 16-31 hold
  // M=lane-16, K={2,3}. Rows >=8 read duplicated data; their outputs are
  // simply never consumed. B layout (4x16 f32) mirrors A in K; packed pairs
  // in phiP make each lane's fetch a single 8-byte load.
  {
    const int m     = lane & 15;
    const int kofs  = (lane >> 4) << 1;          // 0 or 2
    const int ncol2 = (lane & 15) << 1;
    v8f acc0 = {0.f, 0.f, 0.f, 0.f, 0.f, 0.f, 0.f, 0.f};
    v8f acc1 = acc0;
    const float* ap  = xs + (size_t)(m & 7) * NC + kofs + wave * 4;
    const float* bpA = phiP + ncol2      + (size_t)(wave * 4 + kofs) * 32;
    const float* bpB = phiP + ncol2 + 32 + (size_t)(wave * 4 + kofs) * 32;
    #pragma unroll 4
    for (int s = 0; s < NC / (NWAVES * 4); ++s) {          // uniform 256 iters
      v2f a  = *(const v2f*)(ap);
      v2f bA = *(const v2f*)(bpA);
      v2f bB = *(const v2f*)(bpB);
      acc0 = __builtin_amdgcn_wmma_f32_16x16x4_f32(false, a, false, bA,
                                                   (short)0, acc0, false, false);
      acc1 = __builtin_amdgcn_wmma_f32_16x16x4_f32(false, a, false, bB,
                                                   (short)0, acc1, false, false);
      ap  += NWAVES * 4;                 // 32 K per block-step
      bpA += NWAVES * 4 * 32;            // 32 K * 32 packed floats per K
      bpB += NWAVES * 4 * 32;
    }
    // C/D layout: VGPR r, lanes 0-15 -> (M=r, N=lane); lanes 16-31 -> (M=r+8)
    #pragma unroll
    for (int r = 0; r < 8; ++r) {
      int mrow = r + ((lane >> 4) << 3);
      cpart[wave][mrow][(lane & 15)]      = acc0[r];
      cpart[wave][mrow][(lane & 15) + 16] = acc1[r];
    }
  }
  __syncthreads();
  #pragma unroll
  for (int s = 0; s < (16 * PC) / NTHREADS; ++s) {         // cross-wave K reduce
    int o = t + s * NTHREADS;
    int mm = o >> 5, cc = o & 31;
    float sum = 0.f;
    for (int w = 0; w < NWAVES; ++w) sum += cpart[w][mm][cc];
    dots[mm][cc] = sum;
  }
  __syncthreads();

  // ---- Phase 2: gates + Sinkhorn (one lane per row) -----------------------
  if (t < ROWS) {
    const float inv = rms_inv_s[t];              // dot(x_norm,phi) = dot/rms
    const float ap = a_pre[0], apo = a_post[0], ar = a_res[0];
    float M[16];
    #pragma unroll
    for (int i = 0; i < 4; ++i) {
      float zp = ap  * dots[t][i]     * inv + b_pre[i];
      float zq = apo * dots[t][4 + i] * inv + b_post[i];
      Hpre_s [t][i] = 1.f / (1.f + __expf(-zp));
      Hpost_s[t][i] = 2.f / (1.f + __expf(-zq));
    }
    #pragma unroll
    for (int ij = 0; ij < 16; ++ij)
      M[ij] = __expf(ar * dots[t][8 + ij] * inv + b_res[ij]);
    for (int it = 0; it < SINK_IT; ++it) {
      #pragma unroll
      for (int i = 0; i < 4; ++i) {              // rows (axis=-1)
        float q = 1.f / (M[i*4+0] + M[i*4+1] + M[i*4+2] + M[i*4+3] + EPS_F);
        M[i*4+0] *= q; M[i*4+1] *= q; M[i*4+2] *= q; M[i*4+3] *= q;
      }
      #pragma unroll
      for (int j = 0; j < 4; ++j) {              // cols (axis=-2)
        float q = 1.f / (M[0*4+j] + M[1*4+j] + M[2*4+j] + M[3*4+j] + EPS_F);
        M[0*4+j] *= q; M[1*4+j] *= q; M[2*4+j] *= q; M[3*4+j] *= q;
      }
    }
    #pragma unroll
    for (int ij = 0; ij < 16; ++ij) Msink_s[t][ij] = M[ij];
  }
  __syncthreads();

  // ---- Phase 3: aggregated-channel RMS (x_agg recomputed from LDS) --------
  {
    int r = t >> 5;
    const float* xr = xs + (size_t)r * NC + lane;
    float h0 = Hpre_s[r][0], h1 = Hpre_s[r][1],
          h2 = Hpre_s[r][2], h3 = Hpre_s[r][3];
    float acc = 0.f;
    for (int s = 0; s < C_DIM / 32; ++s) {                 // uniform 64 iters
      int c = s * 32;
      float a = h0 * xr[c] + h1 * xr[C_DIM + c]
              + h2 * xr[2 * C_DIM + c] + h3 * xr[3 * C_DIM + c];
      acc = fmaf(a, a, acc);
    }
    redbuf[t] = acc;
  }
  __syncthreads();
  if (t < ROWS) {
    float s = 0.f;
    for (int l = 0; l < 32; ++l) s += redbuf[t * 32 + l];
    agg_inv_s[t] = 1.f / sqrtf(s / (float)C_DIM + EPS_F);
  }
  __syncthreads();

  // ---- Phase 4: out = M @ x + H_post * (x_agg * agg_inv * weight) ---------
  for (int s = 0; s < (ROWS * C_DIM / 4) / NTHREADS; ++s) { // uniform 16 iters
    int q  = t + s * NTHREADS;
    int r  = q >> 9;                    // 512 float4 groups per row
    int c4 = (q & 511) << 2;
    const float* xr = xs + (size_t)r * NC;
    float4 v0 = *(const float4*)(xr + c4);
    float4 v1 = *(const float4*)(xr + C_DIM + c4);
    float4 v2 = *(const float4*)(xr + 2 * C_DIM + c4);
    float4 v3 = *(const float4*)(xr + 3 * C_DIM + c4);
    float4 wq = *(const float4*)(wvec + c4);
    float X0[4] = {v0.x, v0.y, v0.z, v0.w};
    float X1[4] = {v1.x, v1.y, v1.z, v1.w};
    float X2[4] = {v2.x, v2.y, v2.z, v2.w};
    float X3[4] = {v3.x, v3.y, v3.z, v3.w};
    float W4[4] = {wq.x, wq.y, wq.z, wq.w};
    float hp[4], hq[4], Mr[16];
    #pragma unroll
    for (int i = 0; i < 4; ++i) { hp[i] = Hpre_s[r][i]; hq[i] = Hpost_s[r][i]; }
    #pragma unroll
    for (int ij = 0; ij < 16; ++ij) Mr[ij] = Msink_s[r][ij];
    float ai = agg_inv_s[r];
    float o[4][4];
    #pragma unroll
    for (int cc = 0; cc < 4; ++cc) {
      float xagg = hp[0]*X0[cc] + hp[1]*X1[cc] + hp[2]*X2[cc] + hp[3]*X3[cc];
      float xn   = xagg * ai * W4[cc];
      #pragma unroll
      for (int i = 0; i < 4; ++i) {
        o[i][cc] = Mr[i*4+0]*X0[cc] + Mr[i*4+1]*X1[cc]
                 + Mr[i*4+2]*X2[cc] + Mr[i*4+3]*X3[cc] + hq[i]*xn;
      }
    }
    size_t obase = ((size_t)(b0 + r)) * NC + c4;
    #pragma unroll
    for (int i = 0; i < 4; ++i) {
      *(float4*)(outp + obase + (size_t)i * C_DIM) =
          make_float4(o[i][0], o[i][1], o[i][2], o[i][3]);
    }
  }
}

extern "C" void kernel_launch(void* const* d_in, const int* in_sizes, int n_in,
                              void* d_out, int out_size, void* d_ws, size_t ws_size,
                              hipStream_t stream) {
  (void)in_sizes; (void)n_in; (void)out_size; (void)ws_size;
  const float* x        = (const float*)d_in[0];
  const float* wvec     = (const float*)d_in[1];
  const float* phi_pre  = (const float*)d_in[2];
  const float* phi_post = (const float*)d_in[3];
  const float* phi_res  = (const float*)d_in[4];
  const float* b_pre    = (const float*)d_in[5];
  const float* b_post   = (const float*)d_in[6];
  const float* b_res    = (const float*)d_in[7];
  const float* a_pre    = (const float*)d_in[8];
  const float* a_post   = (const float*)d_in[9];
  const float* a_res    = (const float*)d_in[10];
  float* phiP = (float*)d_ws;                       // 8192*32*4 = 1 MB scratch
  float* outp = (float*)d_out;

  pack_phi_kernel<<<(NC * PC) / NTHREADS, NTHREADS, 0, stream>>>(
      phi_pre, phi_post, phi_res, phiP);
  mhc_fused_kernel<<<B_TOT / ROWS, NTHREADS, 0, stream>>>(
      x, wvec, phiP, b_pre, b_post, b_res, a_pre, a_post, a_res, outp);
}